// SVT_channel_token_mixing_65180423684943
// MI455X (gfx1250) — compile-verified
//
#include <hip/hip_runtime.h>

// ---------------------------------------------------------------------------
// DTCWT channel/token mixing for MI455X (gfx1250, wave32).
//   - Channel mixing (4 blocks of 16x16 complex matmuls) and token mixing
//     (14x14 complex matmuls padded to 16) run on V_WMMA_F32_16X16X4_F32,
//     keeping full fp32 precision (C/D = v8f accumulators).
//   - Separable 13/19-tap analysis/synthesis filters are memory-bound VALU
//     kernels with symmetric padding resolved via index folding.
// Workspace requirement: 4*19267584 + 2*12845056 floats = 411,041,792 bytes.
// ---------------------------------------------------------------------------

typedef __attribute__((ext_vector_type(2))) float v2f;
typedef __attribute__((ext_vector_type(8))) float v8f;

#define B_   256
#define H_   28
#define W_   28
#define C_   128
#define CH_  64
#define H2_  14
#define W2_  14
#define N_   (H_ * W_)
#define INVS 0.70710678118654752f

__constant__ float c_H0[13] = {
    -0.0017578f, 0.0f, 0.0222656f, -0.046875f, -0.0482422f, 0.296875f,
    0.5554688f, 0.296875f, -0.0482422f, -0.046875f, 0.0222656f, 0.0f,
    -0.0017578f};
__constant__ float c_H1[19] = {
    -7.06e-05f, 0.0f, 0.0013419f, -0.0018834f, -0.0071568f, 0.023856f,
    0.0556431f, -0.0516881f, -0.2997576f, 0.5594308f, -0.2997576f,
    -0.0516881f, 0.0556431f, 0.023856f, -0.0071568f, -0.0018834f,
    0.0013419f, 0.0f, -7.06e-05f};

// symmetric padding fold (jnp.pad mode='symmetric'): -1 -> 0, n -> n-1
__device__ __forceinline__ int symi(int i, int n) {
  if (i < 0) i = -1 - i;
  if (i >= n) i = 2 * n - 1 - i;
  return i;
}
// g0 = h1 * sgn19 : +h1 at odd taps, -h1 at even taps
__device__ __forceinline__ float g0f(int k) { return (k & 1) ? c_H1[k] : -c_H1[k]; }
// g1 = -(h0 * sgn13): +h0 at odd taps, -h0 at even taps
__device__ __forceinline__ float g1f(int k) { return (k & 1) ? c_H0[k] : -c_H0[k]; }

// fp32 WMMA: D(16x16) += A(16x4) * B(4x16), one K=4 step.
static __device__ __forceinline__ v8f wmma4(v2f a, v2f b, v8f c) {
  return __builtin_amdgcn_wmma_f32_16x16x4_f32(false, a, false, b, (short)0, c,
                                               false, false);
}

// ---------------------------------------------------------------------------
// Stage 1: depthwise 3x3 conv (SAME, zero pad) on channels 0..63 -> out[:, :, 0:64]
// ---------------------------------------------------------------------------
__global__ void k_dwconv(const float* __restrict__ x, const float* __restrict__ w,
                         const float* __restrict__ bias, float* __restrict__ out) {
  int t = blockIdx.x * blockDim.x + threadIdx.x;
  if (t >= B_ * CH_ * N_) return;
  int wc = t % W_;
  int hr = (t / W_) % H_;
  int c = (t / N_) % CH_;
  int b = t / (N_ * CH_);
  float acc = bias[c];
#pragma unroll
  for (int dy = 0; dy < 3; ++dy) {
    int hh = hr + dy - 1;
    if (hh < 0 || hh >= H_) continue;
#pragma unroll
    for (int dx = 0; dx < 3; ++dx) {
      int ww = wc + dx - 1;
      if (ww < 0 || ww >= W_) continue;
      acc += x[(b * N_ + hh * W_ + ww) * C_ + c] * w[c * 9 + dy * 3 + dx];
    }
  }
  out[(b * N_ + hr * W_ + wc) * C_ + c] = acc;
}

// ---------------------------------------------------------------------------
// Stage 2: forward row filters on x2 (channels 64..127): lo = x*h0, hi = x*h1
// ---------------------------------------------------------------------------
__global__ void k_rowfilt(const float* __restrict__ x, float* __restrict__ lo,
                          float* __restrict__ hi) {
  int t = blockIdx.x * blockDim.x + threadIdx.x;
  if (t >= B_ * CH_ * N_) return;
  int wc = t % W_;
  int hr = (t / W_) % H_;
  int c = (t / N_) % CH_;
  int b = t / (N_ * CH_);
  const float* xr = x + (b * N_ + hr * W_) * C_ + CH_ + c;  // stride C_ per w
  float l = 0.f, h = 0.f;
#pragma unroll
  for (int k = 0; k < 13; ++k) l += c_H0[k] * xr[symi(wc + k - 6, W_) * C_];
#pragma unroll
  for (int k = 0; k < 19; ++k) h += c_H1[k] * xr[symi(wc + k - 9, W_) * C_];
  int o = (b * CH_ + c) * N_ + hr * W_ + wc;
  lo[o] = l;
  hi[o] = h;
}

// ---------------------------------------------------------------------------
// Stage 3: ll = colfilt(lo, h0);  xl = ll * w_ll
// ---------------------------------------------------------------------------
__global__ void k_colll(const float* __restrict__ lo, const float* __restrict__ wll,
                        float* __restrict__ xl) {
  int t = blockIdx.x * blockDim.x + threadIdx.x;
  if (t >= B_ * CH_ * N_) return;
  int wc = t % W_;
  int hr = (t / W_) % H_;
  int c = (t / N_) % CH_;
  int b = t / (N_ * CH_);
  const float* col = lo + (b * CH_ + c) * N_ + wc;
  float acc = 0.f;
#pragma unroll
  for (int k = 0; k < 13; ++k) acc += c_H0[k] * col[symi(hr + k - 6, H_) * W_];
  xl[(b * CH_ + c) * N_ + hr * W_ + wc] = acc * wll[c * N_ + hr * W_ + wc];
}

// ---------------------------------------------------------------------------
// Stage 4: column filters lh/hl/hh evaluated only on the 2x2 quads + q2c.
// Band layout (channel-mix friendly): Xr/Xi[pos*64 + c],
//   pos = ((b*6 + s)*14 + i)*14 + j ; s in {0:lhp,1:hhp,2:hlp,3:hlq,4:hhq,5:lhq}
// ---------------------------------------------------------------------------
__global__ void k_q2c_fwd(const float* __restrict__ lo, const float* __restrict__ hi,
                          float* __restrict__ Xr, float* __restrict__ Xi) {
  int t = blockIdx.x * blockDim.x + threadIdx.x;
  if (t >= B_ * CH_ * H2_ * W2_) return;
  int j = t % W2_;
  int i = (t / W2_) % H2_;
  int c = (t / (W2_ * H2_)) % CH_;
  int b = t / (W2_ * H2_ * CH_);
  const float* lc = lo + (b * CH_ + c) * N_ + 2 * j;
  const float* hc = hi + (b * CH_ + c) * N_ + 2 * j;
  float lh[2][2], hl[2][2], hh[2][2];
#pragma unroll
  for (int dy = 0; dy < 2; ++dy) {
#pragma unroll
    for (int dx = 0; dx < 2; ++dx) {
      int y = 2 * i + dy;
      float a1 = 0.f, a0 = 0.f, a2 = 0.f;
#pragma unroll
      for (int k = 0; k < 19; ++k) {
        int ys = symi(y + k - 9, H_) * W_;
        a1 += c_H1[k] * lc[ys + dx];
        a2 += c_H1[k] * hc[ys + dx];
      }
#pragma unroll
      for (int k = 0; k < 13; ++k) {
        int ys = symi(y + k - 6, H_) * W_;
        a0 += c_H0[k] * hc[ys + dx];
      }
      lh[dy][dx] = a1 * INVS;
      hl[dy][dx] = a0 * INVS;
      hh[dy][dx] = a2 * INVS;
    }
  }
  int base = b * 6 * H2_ * W2_ + i * W2_ + j;
  auto store = [&](int s1, int s2, float q[2][2]) {
    float a = q[0][0], bb = q[0][1], cc = q[1][0], d = q[1][1];
    int p1 = (base + s1 * H2_ * W2_) * CH_ + c;
    int p2 = (base + s2 * H2_ * W2_) * CH_ + c;
    Xr[p1] = a - d;
    Xi[p1] = bb + cc;
    Xr[p2] = a + d;
    Xi[p2] = bb - cc;
  };
  store(0, 5, lh);
  store(2, 3, hl);
  store(1, 4, hh);
}

// ---------------------------------------------------------------------------
// Stage 5: channel mixing. One wave = 16 positions x all 4 channel blocks.
// Complex 2-layer block-diag matmul on WMMA fp32 16x16x4.
// A-layout: lane(0..15) holds row M, VGPRs = K pair; lanes 16..31 hold K+2.
// Output written in token-mix layout: T[(u)*196 + h2*14 + w2], u=(b*64+c)*6+s.
// ---------------------------------------------------------------------------
__global__ void __launch_bounds__(256) k_chanmix(
    const float* __restrict__ Xr, const float* __restrict__ Xi,
    const float* __restrict__ w1, const float* __restrict__ w2,
    const float* __restrict__ b1, const float* __restrict__ b2,
    float* __restrict__ Tr, float* __restrict__ Ti) {
  __shared__ float lds[8][2][256];
  const int lane = threadIdx.x & 31;
  const int wv = threadIdx.x >> 5;
  const int row = lane & 15;
  const int hil = lane >> 4;  // 0/1
  const int p0 = (blockIdx.x * 8 + wv) * 16;
  float* Lr = &lds[wv][0][0];
  float* Li = &lds[wv][1][0];

  for (int kb = 0; kb < 4; ++kb) {
    // ---- A tiles (16x16 as 4 K-chunks of 4), fp32, vector b64 loads
    v2f ar[4], ai[4];
#pragma unroll
    for (int q = 0; q < 4; ++q) {
      int kk = 4 * q + 2 * hil;
      int idx = (p0 + row) * CH_ + kb * 16 + kk;
      ar[q] = *reinterpret_cast<const v2f*>(&Xr[idx]);
      ai[q] = *reinterpret_cast<const v2f*>(&Xi[idx]);
    }
    // ---- B tiles for layer 1: W0 = w_lh1[0][kb], W1 = w_lh1[1][kb]
    const float* W0 = w1 + kb * 256;
    const float* W1 = w1 + 1024 + kb * 256;
    v2f B0[4], B1[4], B1n[4];
#pragma unroll
    for (int q = 0; q < 4; ++q) {
      int d0 = 4 * q + hil;
      v2f t0 = {W0[d0 * 16 + row], W0[(d0 + 2) * 16 + row]};
      v2f t1 = {W1[d0 * 16 + row], W1[(d0 + 2) * 16 + row]};
      B0[q] = t0;
      B1[q] = t1;
      B1n[q] = -t1;
    }
    float br = b1[kb * 16 + row];
    float bi = b1[64 + kb * 16 + row];
    v8f accr, acci;
#pragma unroll
    for (int r = 0; r < 8; ++r) { accr[r] = br; acci[r] = bi; }
    // xr1 = relu(xr*W0 - xi*W1 + b0) ; xi1 = relu(xr*W1 + xi*W0 + b1)
#pragma unroll
    for (int q = 0; q < 4; ++q) accr = wmma4(ar[q], B0[q], accr);
#pragma unroll
    for (int q = 0; q < 4; ++q) accr = wmma4(ai[q], B1n[q], accr);
#pragma unroll
    for (int q = 0; q < 4; ++q) acci = wmma4(ar[q], B1[q], acci);
#pragma unroll
    for (int q = 0; q < 4; ++q) acci = wmma4(ai[q], B0[q], acci);
#pragma unroll
    for (int r = 0; r < 8; ++r) {
      accr[r] = fmaxf(accr[r], 0.f);
      acci[r] = fmaxf(acci[r], 0.f);
    }
    // ---- D-layout -> LDS -> A-layout for layer 2
#pragma unroll
    for (int r = 0; r < 8; ++r) {
      int m = r + 8 * hil;
      Lr[m * 16 + row] = accr[r];
      Li[m * 16 + row] = acci[r];
    }
    __syncthreads();
    v2f cr[4], ci[4];
#pragma unroll
    for (int q = 0; q < 4; ++q) {
      int kk = 4 * q + 2 * hil;
      cr[q] = *reinterpret_cast<const v2f*>(&Lr[row * 16 + kk]);
      ci[q] = *reinterpret_cast<const v2f*>(&Li[row * 16 + kk]);
    }
    __syncthreads();
    // ---- layer 2
    const float* V0 = w2 + kb * 256;
    const float* V1 = w2 + 1024 + kb * 256;
    v2f C0[4], C1[4], C1n[4];
#pragma unroll
    for (int q = 0; q < 4; ++q) {
      int d0 = 4 * q + hil;
      v2f t0 = {V0[d0 * 16 + row], V0[(d0 + 2) * 16 + row]};
      v2f t1 = {V1[d0 * 16 + row], V1[(d0 + 2) * 16 + row]};
      C0[q] = t0;
      C1[q] = t1;
      C1n[q] = -t1;
    }
    float br2 = b2[kb * 16 + row];
    float bi2 = b2[64 + kb * 16 + row];
    v8f dr, di;
#pragma unroll
    for (int r = 0; r < 8; ++r) { dr[r] = br2; di[r] = bi2; }
#pragma unroll
    for (int q = 0; q < 4; ++q) dr = wmma4(cr[q], C0[q], dr);
#pragma unroll
    for (int q = 0; q < 4; ++q) dr = wmma4(ci[q], C1n[q], dr);
#pragma unroll
    for (int q = 0; q < 4; ++q) di = wmma4(cr[q], C1[q], di);
#pragma unroll
    for (int q = 0; q < 4; ++q) di = wmma4(ci[q], C0[q], di);
    // ---- scatter to token layout
    int cch = kb * 16 + row;
#pragma unroll
    for (int r = 0; r < 8; ++r) {
      int p = p0 + r + 8 * hil;
      int j = p % 14;
      int tmp = p / 14;
      int ii = tmp % 14;
      tmp /= 14;
      int s = tmp % 6;
      int b = tmp / 6;
      int o = (((b * CH_ + cch) * 6 + s) * H2_ + ii) * W2_ + j;
      Tr[o] = dr[r];
      Ti[o] = di[r];
    }
  }
}

// ---------------------------------------------------------------------------
// Stage 6: token mixing. One wave = 16 rows u=(b*64+c)*6+s at one h2.
// 14x14 complex matmul zero-padded to 16x16; padded lanes never contaminate.
// ---------------------------------------------------------------------------
__global__ void __launch_bounds__(256) k_tokenmix(
    const float* __restrict__ Tr, const float* __restrict__ Ti,
    const float* __restrict__ wt1, const float* __restrict__ wt2,
    const float* __restrict__ bt1, const float* __restrict__ bt2,
    float* __restrict__ Or, float* __restrict__ Oi) {
  __shared__ float lds[8][2][256];
  const int lane = threadIdx.x & 31;
  const int wv = threadIdx.x >> 5;
  const int row = lane & 15;
  const int hil = lane >> 4;
  const int tile = blockIdx.x * 8 + wv;
  const int h2 = tile / 6144;
  const int u0 = (tile % 6144) * 16;
  float* Lr = &lds[wv][0][0];
  float* Li = &lds[wv][1][0];

  const float* U0 = wt1 + h2 * 196;
  const float* U1 = wt1 + 2744 + h2 * 196;
  const float* V0 = wt2 + h2 * 196;
  const float* V1 = wt2 + 2744 + h2 * 196;

  // ---- A tiles (zero-padded columns 14,15)
  v2f ar[4], ai[4];
#pragma unroll
  for (int q = 0; q < 4; ++q) {
    int kk = 4 * q + 2 * hil;
    int base = (u0 + row) * 196 + h2 * 14;
    v2f tr, ti;
    tr.x = (kk < 14) ? Tr[base + kk] : 0.f;
    tr.y = (kk + 1 < 14) ? Tr[base + kk + 1] : 0.f;
    ti.x = (kk < 14) ? Ti[base + kk] : 0.f;
    ti.y = (kk + 1 < 14) ? Ti[base + kk + 1] : 0.f;
    ar[q] = tr;
    ai[q] = ti;
  }
  auto loadB = [&](const float* Wm, v2f* Bm) {
#pragma unroll
    for (int q = 0; q < 4; ++q) {
      int d0 = 4 * q + hil;
      v2f t;
      t.x = (d0 < 14 && row < 14) ? Wm[d0 * 14 + row] : 0.f;
      t.y = (d0 + 2 < 14 && row < 14) ? Wm[(d0 + 2) * 14 + row] : 0.f;
      Bm[q] = t;
    }
  };
  v2f B0[4], B1[4], B1n[4];
  loadB(U0, B0);
  loadB(U1, B1);
#pragma unroll
  for (int q = 0; q < 4; ++q) B1n[q] = -B1[q];

  float br = (row < 14) ? bt1[h2 * 14 + row] : 0.f;
  float bi = (row < 14) ? bt1[196 + h2 * 14 + row] : 0.f;
  v8f accr, acci;
#pragma unroll
  for (int r = 0; r < 8; ++r) { accr[r] = br; acci[r] = bi; }
#pragma unroll
  for (int q = 0; q < 4; ++q) accr = wmma4(ar[q], B0[q], accr);
#pragma unroll
  for (int q = 0; q < 4; ++q) accr = wmma4(ai[q], B1n[q], accr);
#pragma unroll
  for (int q = 0; q < 4; ++q) acci = wmma4(ar[q], B1[q], acci);
#pragma unroll
  for (int q = 0; q < 4; ++q) acci = wmma4(ai[q], B0[q], acci);
#pragma unroll
  for (int r = 0; r < 8; ++r) {
    accr[r] = fmaxf(accr[r], 0.f);
    acci[r] = fmaxf(acci[r], 0.f);
  }
#pragma unroll
  for (int r = 0; r < 8; ++r) {
    int m = r + 8 * hil;
    Lr[m * 16 + row] = accr[r];
    Li[m * 16 + row] = acci[r];
  }
  __syncthreads();
  v2f cr[4], ci[4];
#pragma unroll
  for (int q = 0; q < 4; ++q) {
    int kk = 4 * q + 2 * hil;
    cr[q] = *reinterpret_cast<const v2f*>(&Lr[row * 16 + kk]);
    ci[q] = *reinterpret_cast<const v2f*>(&Li[row * 16 + kk]);
  }
  __syncthreads();
  v2f C0[4], C1[4], C1n[4];
  loadB(V0, C0);
  loadB(V1, C1);
#pragma unroll
  for (int q = 0; q < 4; ++q) C1n[q] = -C1[q];
  float br2 = (row < 14) ? bt2[h2 * 14 + row] : 0.f;
  float bi2 = (row < 14) ? bt2[196 + h2 * 14 + row] : 0.f;
  v8f dr, di;
#pragma unroll
  for (int r = 0; r < 8; ++r) { dr[r] = br2; di[r] = bi2; }
#pragma unroll
  for (int q = 0; q < 4; ++q) dr = wmma4(cr[q], C0[q], dr);
#pragma unroll
  for (int q = 0; q < 4; ++q) dr = wmma4(ci[q], C1n[q], dr);
#pragma unroll
  for (int q = 0; q < 4; ++q) di = wmma4(cr[q], C1[q], di);
#pragma unroll
  for (int q = 0; q < 4; ++q) di = wmma4(ci[q], C0[q], di);
  if (row < 14) {
#pragma unroll
    for (int r = 0; r < 8; ++r) {
      int u = u0 + r + 8 * hil;
      int o = u * 196 + h2 * 14 + row;
      Or[o] = dr[r];
      Oi[o] = di[r];
    }
  }
}

// ---------------------------------------------------------------------------
// Stage 7: c2q — rebuild lh/hl/hh (B,64,28,28) from band pairs (0,5)(2,3)(1,4)
// ---------------------------------------------------------------------------
__global__ void k_c2q(const float* __restrict__ Or, const float* __restrict__ Oi,
                      float* __restrict__ LH, float* __restrict__ HL,
                      float* __restrict__ HH) {
  int t = blockIdx.x * blockDim.x + threadIdx.x;
  if (t >= B_ * CH_ * H2_ * W2_) return;
  int j = t % W2_;
  int i = (t / W2_) % H2_;
  int c = (t / (W2_ * H2_)) % CH_;
  int b = t / (W2_ * H2_ * CH_);
  int ub = (b * CH_ + c) * 6;
  int off = i * 14 + j;
  auto emit = [&](int s1, int s2, float* dst) {
    int o1 = (ub + s1) * 196 + off;
    int o2 = (ub + s2) * 196 + off;
    float w1r = Or[o1], w1i = Oi[o1], w2r = Or[o2], w2i = Oi[o2];
    int ob = (b * CH_ + c) * N_ + (2 * i) * W_ + 2 * j;
    dst[ob] = (w1r + w2r) * INVS;
    dst[ob + 1] = (w1i + w2i) * INVS;
    dst[ob + W_] = (w1i - w2i) * INVS;
    dst[ob + W_ + 1] = (w2r - w1r) * INVS;
  };
  emit(0, 5, LH);
  emit(2, 3, HL);
  emit(1, 4, HH);
}

// ---------------------------------------------------------------------------
// Stage 8: inverse column filters: lo2 = xl*g0 + lh*g1 ; hi2 = hl*g0 + hh*g1
// ---------------------------------------------------------------------------
__global__ void k_colinv(const float* __restrict__ XL, const float* __restrict__ LH,
                         const float* __restrict__ HL, const float* __restrict__ HH,
                         float* __restrict__ LO, float* __restrict__ HI) {
  int t = blockIdx.x * blockDim.x + threadIdx.x;
  if (t >= B_ * CH_ * N_) return;
  int wc = t % W_;
  int hr = (t / W_) % H_;
  int c = (t / N_) % CH_;
  int b = t / (N_ * CH_);
  int base = (b * CH_ + c) * N_ + wc;
  float l = 0.f, h = 0.f;
#pragma unroll
  for (int k = 0; k < 19; ++k) {
    int ys = symi(hr + k - 9, H_) * W_;
    float g = g0f(k);
    l += g * XL[base + ys];
    h += g * HL[base + ys];
  }
#pragma unroll
  for (int k = 0; k < 13; ++k) {
    int ys = symi(hr + k - 6, H_) * W_;
    float g = g1f(k);
    l += g * LH[base + ys];
    h += g * HH[base + ys];
  }
  int o = (b * CH_ + c) * N_ + hr * W_ + wc;
  LO[o] = l;
  HI[o] = h;
}

// ---------------------------------------------------------------------------
// Stage 9: inverse row filters -> out[:, :, 64:128]
// ---------------------------------------------------------------------------
__global__ void k_rowinv(const float* __restrict__ LO, const float* __restrict__ HI,
                         float* __restrict__ out) {
  int t = blockIdx.x * blockDim.x + threadIdx.x;
  if (t >= B_ * CH_ * N_) return;
  int wc = t % W_;
  int hr = (t / W_) % H_;
  int c = (t / N_) % CH_;
  int b = t / (N_ * CH_);
  const float* lr = LO + (b * CH_ + c) * N_ + hr * W_;
  const float* hr_ = HI + (b * CH_ + c) * N_ + hr * W_;
  float acc = 0.f;
#pragma unroll
  for (int k = 0; k < 19; ++k) acc += g0f(k) * lr[symi(wc + k - 9, W_)];
#pragma unroll
  for (int k = 0; k < 13; ++k) acc += g1f(k) * hr_[symi(wc + k - 6, W_)];
  out[(b * N_ + hr * W_ + wc) * C_ + CH_ + c] = acc;
}

// ---------------------------------------------------------------------------
extern "C" void kernel_launch(void* const* d_in, const int* in_sizes, int n_in,
                              void* d_out, int out_size, void* d_ws, size_t ws_size,
                              hipStream_t stream) {
  (void)in_sizes; (void)n_in; (void)out_size; (void)ws_size;
  const float* x      = (const float*)d_in[0];
  const float* conv_w = (const float*)d_in[1];
  const float* conv_b = (const float*)d_in[2];
  const float* w_ll   = (const float*)d_in[3];
  const float* w_lh1  = (const float*)d_in[4];
  const float* w_lh2  = (const float*)d_in[5];
  const float* b_lh1  = (const float*)d_in[6];
  const float* b_lh2  = (const float*)d_in[7];
  const float* w_t1   = (const float*)d_in[8];
  const float* w_t2   = (const float*)d_in[9];
  const float* b_t1   = (const float*)d_in[10];
  const float* b_t2   = (const float*)d_in[11];
  float* out = (float*)d_out;
  float* ws = (float*)d_ws;

  const size_t RB = 19267584;  // band region: B*6*14*14*64 floats (77 MB)
  const size_t RS = 12845056;  // image region: B*64*28*28 floats (51 MB)
  float* R0 = ws;               // xr bands -> token out (r) -> lo2
  float* R1 = ws + RB;          // xi bands -> token out (i) -> hi2
  float* R2 = ws + 2 * RB;      // fwd lo   -> txr          -> lh
  float* R3 = ws + 3 * RB;      // fwd hi   -> txi          -> hl
  float* R4 = ws + 4 * RB;      // xl (live stages 3..8)
  float* R5 = ws + 4 * RB + RS; // hh

  dim3 blk(256);
  const int nimg = B_ * CH_ * N_;          // 12,845,056
  const int nq = B_ * CH_ * H2_ * W2_;     // 3,211,264

  k_dwconv <<<(nimg + 255) / 256, blk, 0, stream>>>(x, conv_w, conv_b, out);
  k_rowfilt<<<(nimg + 255) / 256, blk, 0, stream>>>(x, R2, R3);
  k_colll  <<<(nimg + 255) / 256, blk, 0, stream>>>(R2, w_ll, R4);
  k_q2c_fwd<<<(nq + 255) / 256, blk, 0, stream>>>(R2, R3, R0, R1);
  k_chanmix<<<2352, blk, 0, stream>>>(R0, R1, w_lh1, w_lh2, b_lh1, b_lh2, R2, R3);
  k_tokenmix<<<10752, blk, 0, stream>>>(R2, R3, w_t1, w_t2, b_t1, b_t2, R0, R1);
  k_c2q    <<<(nq + 255) / 256, blk, 0, stream>>>(R0, R1, R2, R3, R5);
  k_colinv <<<(nimg + 255) / 256, blk, 0, stream>>>(R4, R2, R3, R5, R0, R1);
  k_rowinv <<<(nimg + 255) / 256, blk, 0, stream>>>(R0, R1, out);
}